// EqualizationTransform_85916525789590
// MI455X (gfx1250) — compile-verified
//
#include <hip/hip_runtime.h>
#include <math.h>

#define PIX (1024 * 1024)
#define NBINS 256
// Per block: 256 threads x 16 pixels = 4096 pixels -> 256 blocks per 1Mpix image.
#define PIX_PER_BLOCK 4096

typedef __attribute__((ext_vector_type(2))) float v2f;
typedef __attribute__((ext_vector_type(8))) float v8f;

__device__ __forceinline__ float clamp01(float x) {
    return fminf(fmaxf(x, 0.0f), 1.0f);
}

// ---------------------------------------------------------------------------
// Pass 0: zero per-image histograms (ws is poisoned; must re-init every call)
// ---------------------------------------------------------------------------
__global__ void zero_hist_kernel(unsigned int* __restrict__ hist) {
    hist[blockIdx.x * NBINS + threadIdx.x] = 0u;
}

// ---------------------------------------------------------------------------
// Pass A: per-image 256-bin histogram of Y (LDS-privatized)
// ---------------------------------------------------------------------------
__global__ void hist_kernel(const float* __restrict__ img,
                            unsigned int* __restrict__ hist) {
    __shared__ unsigned int lh[NBINS];
    const int tid = threadIdx.x;
    lh[tid] = 0u;
    __syncthreads();

    const int n = blockIdx.y;
    const float* rp = img + (size_t)n * 3 * PIX;
    const float* gp = rp + PIX;
    const float* bp = gp + PIX;
    const int base = blockIdx.x * PIX_PER_BLOCK;

    for (int q = 0; q < 4; ++q) {
        const int idx = base + q * 1024 + tid * 4;
        const float4 r4 = *(const float4*)(rp + idx);
        const float4 g4 = *(const float4*)(gp + idx);
        const float4 b4 = *(const float4*)(bp + idx);
        const float rr[4] = {r4.x, r4.y, r4.z, r4.w};
        const float gg[4] = {g4.x, g4.y, g4.z, g4.w};
        const float bb[4] = {b4.x, b4.y, b4.z, b4.w};
#pragma unroll
        for (int k = 0; k < 4; ++k) {
            const float r = clamp01(rr[k]);
            const float g = clamp01(gg[k]);
            const float b = clamp01(bb[k]);
            const float y = 0.299f * r + 0.587f * g + 0.114f * b;
            const float flat = y * 255.0f;
            int bin = (int)floorf(flat * (256.0f / 255.0f));
            bin = min(max(bin, 0), NBINS - 1);
            atomicAdd(&lh[bin], 1u);
        }
    }
    __syncthreads();
    const unsigned int c = lh[tid];
    if (c) atomicAdd(&hist[n * NBINS + tid], c);
}

// ---------------------------------------------------------------------------
// Pass B: build LUT. One wave (32 lanes) per image. The 256-bin inclusive
// prefix sum is computed as a 16x16x16 matmul P = H * U (U = upper-triangular
// ones) using four chained V_WMMA_F32_16X16X4_F32 (K-slices of 4), which is
// exact in f32 since all counts are integers < 2^24. A 16-row offset scan
// then yields cumsum[16*i + k] = P[i][k] + offset[i].
// ---------------------------------------------------------------------------
__global__ void lut_kernel(const unsigned int* __restrict__ hist,
                           float* __restrict__ lut,
                           float* __restrict__ stepflag) {
    __shared__ float hf[NBINS];   // histogram as float
    __shared__ float P[NBINS];    // row-wise inclusive prefix (row-major)
    __shared__ float off[16];     // per-row offsets
    __shared__ int   redi[32];
    __shared__ int   s_last;

    const int lane = threadIdx.x;  // 0..31, one wave, EXEC all ones for WMMA
    const int n = blockIdx.x;
    const unsigned int* h = hist + n * NBINS;

#pragma unroll
    for (int q = 0; q < 8; ++q) hf[lane + 32 * q] = (float)h[lane + 32 * q];
    __syncthreads();

    // A-matrix (16x4 f32) layout: lanes 0-15 hold M=lane, K slots {0,1};
    // lanes 16-31 hold M=lane-16, K slots {2,3}.  B (4x16) mirrors this.
    const int m  = lane & 15;
    const int kb = (lane < 16) ? 0 : 2;

    v8f c = {};  // 16x16 f32 accumulator (zero-initialized C)
#pragma unroll
    for (int t = 0; t < 4; ++t) {
        const int k0 = 4 * t + kb;  // absolute K row of this slice
        v2f a, b;
        a.x = hf[m * 16 + k0];
        a.y = hf[m * 16 + k0 + 1];
        // U[j][col] = 1 if j <= col (col == lane&15 for B layout)
        b.x = (k0 <= m) ? 1.0f : 0.0f;
        b.y = (k0 + 1 <= m) ? 1.0f : 0.0f;
        c = __builtin_amdgcn_wmma_f32_16x16x4_f32(
                /*neg_a=*/false, a, /*neg_b=*/false, b,
                /*c_mod=*/(short)0, c, /*reuse_a=*/false, /*reuse_b=*/false);
    }

    // C/D layout: VGPR v holds row (v + 8*(lane>=16)), column = lane&15.
    const int ncol  = lane & 15;
    const int mrow0 = (lane < 16) ? 0 : 8;
#pragma unroll
    for (int v = 0; v < 8; ++v) P[(mrow0 + v) * 16 + ncol] = c[v];
    __syncthreads();

    // Row offsets: offset[i] = sum of full-row totals of rows < i.
    if (lane < 16) {
        float s = 0.0f;
        for (int r = 0; r < lane; ++r) s += P[r * 16 + 15];
        off[lane] = s;
    }

    // Last nonzero bin (reference: 255 - argmax(reversed nz)).
    int best = -1;
#pragma unroll
    for (int q = 0; q < 8; ++q) {
        const int j = lane + 32 * q;
        if (hf[j] > 0.0f) best = max(best, j);
    }
    redi[lane] = best;
    __syncthreads();
    if (lane == 0) {
        int bb = -1;
        for (int i = 0; i < 32; ++i) bb = max(bb, redi[i]);
        s_last = (bb < 0) ? (NBINS - 1) : bb;
    }
    __syncthreads();

    const int last_idx = s_last;
    const float last_val = hf[last_idx];
    const float total = P[255] + off[15];  // cumsum[255] == sum(histo)
    const float step = floorf((total - last_val) / 255.0f);
    const float safe = fmaxf(step, 1.0f);
    const float hstep = floorf(step * 0.5f);

    // lut[0] = 0; lut[j] = clip(floor((cumsum[j-1] + floor(step/2))/safe), 0, 255)
#pragma unroll
    for (int q = 0; q < 8; ++q) {
        const int j = lane + 32 * q;
        float l = 0.0f;
        if (j > 0) {
            const int jm = j - 1;
            const float cs = P[jm] + off[jm >> 4];
            l = floorf((cs + hstep) / safe);
            l = fminf(fmaxf(l, 0.0f), 255.0f);
        }
        lut[n * NBINS + j] = l;
    }
    if (lane == 0) stepflag[n] = (step == 0.0f) ? 1.0f : 0.0f;
}

// ---------------------------------------------------------------------------
// Pass C: recompute YUV per pixel, remap Y through LUT, emit RGB.
// ---------------------------------------------------------------------------
__global__ void map_kernel(const float* __restrict__ img,
                           const float* __restrict__ lut,
                           const float* __restrict__ stepflag,
                           float* __restrict__ out) {
    __shared__ float llut[NBINS];
    __shared__ float sflag;
    const int tid = threadIdx.x;
    const int n = blockIdx.y;

    llut[tid] = lut[n * NBINS + tid];
    if (tid == 0) sflag = stepflag[n];
    __syncthreads();
    const bool ident = (sflag != 0.0f);

    const float* rp = img + (size_t)n * 3 * PIX;
    const float* gp = rp + PIX;
    const float* bp = gp + PIX;
    float* ro = out + (size_t)n * 3 * PIX;
    float* go = ro + PIX;
    float* bo = go + PIX;
    const int base = blockIdx.x * PIX_PER_BLOCK;

    for (int q = 0; q < 4; ++q) {
        const int idx = base + q * 1024 + tid * 4;
        if (q < 3) {  // prefetch next iteration's cachelines (global_prefetch_b8)
            __builtin_prefetch(rp + idx + 1024, 0, 3);
            __builtin_prefetch(gp + idx + 1024, 0, 3);
            __builtin_prefetch(bp + idx + 1024, 0, 3);
        }
        const float4 r4 = *(const float4*)(rp + idx);
        const float4 g4 = *(const float4*)(gp + idx);
        const float4 b4 = *(const float4*)(bp + idx);
        const float rr[4] = {r4.x, r4.y, r4.z, r4.w};
        const float gg[4] = {g4.x, g4.y, g4.z, g4.w};
        const float bb[4] = {b4.x, b4.y, b4.z, b4.w};
        float orr[4], ogg[4], obb[4];
#pragma unroll
        for (int k = 0; k < 4; ++k) {
            const float r = clamp01(rr[k]);
            const float g = clamp01(gg[k]);
            const float b = clamp01(bb[k]);
            const float y = 0.299f * r + 0.587f * g + 0.114f * b;
            const float u = -0.147f * r - 0.289f * g + 0.436f * b;
            const float v = 0.615f * r - 0.515f * g - 0.1f * b;
            const float flat = y * 255.0f;
            int gi = (int)flat;  // trunc == astype(int32); flat >= 0
            gi = min(max(gi, 0), NBINS - 1);
            const float yo = ident ? flat : llut[gi];
            const float ye = yo / 255.0f;
            orr[k] = ye + 1.14f * v;
            ogg[k] = ye - 0.396f * u - 0.581f * v;
            obb[k] = ye + 2.029f * u;
        }
        *(float4*)(ro + idx) = make_float4(orr[0], orr[1], orr[2], orr[3]);
        *(float4*)(go + idx) = make_float4(ogg[0], ogg[1], ogg[2], ogg[3]);
        *(float4*)(bo + idx) = make_float4(obb[0], obb[1], obb[2], obb[3]);
    }
}

// ---------------------------------------------------------------------------
extern "C" void kernel_launch(void* const* d_in, const int* in_sizes, int n_in,
                              void* d_out, int out_size, void* d_ws, size_t ws_size,
                              hipStream_t stream) {
    const float* img = (const float*)d_in[0];
    float* out = (float*)d_out;

    const int nimg = in_sizes[0] / (3 * PIX);  // 16 for the reference shapes
    const int blocks_per_img = PIX / PIX_PER_BLOCK;  // 256

    unsigned int* hist = (unsigned int*)d_ws;
    float* lut = (float*)((char*)d_ws + (size_t)nimg * NBINS * sizeof(unsigned int));
    float* stepflag = (float*)((char*)lut + (size_t)nimg * NBINS * sizeof(float));

    zero_hist_kernel<<<nimg, NBINS, 0, stream>>>(hist);
    hist_kernel<<<dim3(blocks_per_img, nimg), 256, 0, stream>>>(img, hist);
    lut_kernel<<<nimg, 32, 0, stream>>>(hist, lut, stepflag);
    map_kernel<<<dim3(blocks_per_img, nimg), 256, 0, stream>>>(img, lut, stepflag, out);
}